// HopfieldModel_37417755083724
// MI455X (gfx1250) — compile-verified
//
#include <hip/hip_runtime.h>
#include <hip/hip_bf16.h>
#include <math.h>
#include <stdint.h>

#define BATCH 128
#define IMG 64
#define DD 1024
#define HID 2048
#define MAX_STEPS 16
#define DT0 0.1f
#define TOLV 0.01f
#define MIN_DT 0.1f
#define MAX_DELTA_T 20.0f
#define REL_TOL 1e-3f
#define LN_EPS 1e-3f

typedef float v2f __attribute__((ext_vector_type(2)));
typedef float v8f __attribute__((ext_vector_type(8)));

// Async global->LDS copy (CDNA5): per-lane 16B, tracked by ASYNCcnt.
__device__ __forceinline__ void async_b128(uint32_t lds_off, const float* gp) {
  asm volatile("global_load_async_to_lds_b128 %0, %1, off"
               :: "v"(lds_off), "v"(gp) : "memory");
}
__device__ __forceinline__ void wait_async0() {
  asm volatile("s_wait_asynccnt 0x0" ::: "memory");
}

// ---------------------------------------------------------------------------
// Preprocess: 2x2 maxpool + LayerNorm(ln0) -> x0 [BATCH, DD]
// ---------------------------------------------------------------------------
__global__ __launch_bounds__(256) void k_preprocess(
    const float* __restrict__ xin, const float* __restrict__ g,
    const float* __restrict__ bta, float* __restrict__ x0) {
  __shared__ float ps[DD];
  __shared__ float red[256];
  __shared__ float red2[256];
  int b = blockIdx.x;
  int tid = threadIdx.x;
  const float* src = xin + (size_t)b * IMG * IMG;
  float s = 0.f, s2 = 0.f;
  for (int i = tid; i < DD; i += 256) {
    int r = i >> 5, c = i & 31;
    const float* p0 = src + (2 * r) * IMG + 2 * c;
    float v = fmaxf(fmaxf(p0[0], p0[1]), fmaxf(p0[IMG], p0[IMG + 1]));
    ps[i] = v;
    s += v;
    s2 += v * v;
  }
  red[tid] = s;
  red2[tid] = s2;
  __syncthreads();
  for (int off = 128; off > 0; off >>= 1) {
    if (tid < off) { red[tid] += red[tid + off]; red2[tid] += red2[tid + off]; }
    __syncthreads();
  }
  float mean = red[0] * (1.0f / DD);
  float var = red2[0] * (1.0f / DD) - mean * mean;
  float rstd = rsqrtf(var + LN_EPS);
  for (int i = tid; i < DD; i += 256)
    x0[(size_t)b * DD + i] = (ps[i] - mean) * rstd * g[i] + bta[i];
}

// ---------------------------------------------------------------------------
// ctrl[0]=t, ctrl[1]=dt, ctrl[2]=done, ctrl[3]=upd
// ---------------------------------------------------------------------------
__global__ void k_init(float* ctrl, unsigned* errb, float* prevE) {
  int t = threadIdx.x;
  if (t < BATCH) prevE[t] = 0.f;
  if (t == 0) {
    ctrl[0] = 0.f; ctrl[1] = DT0; ctrl[2] = 0.f; ctrl[3] = 0.f;
    errb[0] = 0u;
  }
}

// ---------------------------------------------------------------------------
// Forward GEMM via V_WMMA_F32_16X16X4_F32, with fused RK stage combination:
//   z = x + dt*(c1 k1 + ... + c5 k5)   (computed at A-tile load)
//   A = (mode==0 ? g_reflect(z) : z) [BATCH, DD], B = W1 [DD, HID]
//   mode 0: out = (A@W1 + b1 > 0) ? W2[col] : 0        (grad seed)
//   mode 1: out = relu(A@W1 + b1)                       (energy acts)
// Block: 256 threads = 8 waves; block tile 16(M) x 128(N); wave tile 16x16.
// ---------------------------------------------------------------------------
__global__ __launch_bounds__(256) void k_fwd(
    const float* __restrict__ x, const float* __restrict__ k1,
    const float* __restrict__ k2, const float* __restrict__ k3,
    const float* __restrict__ k4, const float* __restrict__ k5,
    float c1, float c2, float c3, float c4, float c5,
    const float* __restrict__ ctrl, const float* __restrict__ W1,
    const float* __restrict__ b1, const float* __restrict__ W2,
    float* __restrict__ outp, int mode) {
  __shared__ float As[16][68];
  int nBase = blockIdx.x * 128;
  int mBase = blockIdx.y * 16;
  int tid = threadIdx.x;
  int wave = tid >> 5;
  int lane = tid & 31;
  int lrow = lane & 15;
  int khalf = lane >> 4;
  int col = nBase + wave * 16 + lrow;
  float dt = ctrl[1];
  v8f acc = {};
  for (int k0 = 0; k0 < DD; k0 += 64) {
    __syncthreads();
    for (int idx = tid; idx < 16 * 64; idx += 256) {
      int r = idx >> 6, c = idx & 63;
      size_t gi = (size_t)(mBase + r) * DD + k0 + c;
      float v = x[gi] + dt * (c1 * k1[gi] + c2 * k2[gi] + c3 * k3[gi] +
                              c4 * k4[gi] + c5 * k5[gi]);
      if (mode == 0) {  // g_reflect
        int iv = (int)v;
        v = ((iv & 1) == 0) ? v : 1.0f - v;
      }
      As[r][c] = v;
    }
    __syncthreads();
#pragma unroll
    for (int kk = 0; kk < 16; ++kk) {
      int kb = kk * 4 + 2 * khalf;
      v2f a, bf;
      a.x = As[lrow][kb];
      a.y = As[lrow][kb + 1];
      bf.x = W1[(size_t)(k0 + kb) * HID + col];
      bf.y = W1[(size_t)(k0 + kb + 1) * HID + col];
      acc = __builtin_amdgcn_wmma_f32_16x16x4_f32(false, a, false, bf,
                                                  (short)0, acc, false, false);
    }
  }
  int mOff = mBase + khalf * 8;
  float w2v = W2[col];
  float b1v = b1[col];
#pragma unroll
  for (int r = 0; r < 8; ++r) {
    float val = acc[r] + b1v;
    float o = (mode == 0) ? ((val > 0.f) ? w2v : 0.f) : fmaxf(val, 0.f);
    outp[(size_t)(mOff + r) * HID + col] = o;
  }
}

// ---------------------------------------------------------------------------
// Backward GEMM: k_s = -( D[BATCH,HID] @ W1^T ) * reflect_sign(z)
// B[k,n] = W1[n,k]; the W1^T tile is staged into LDS (k-major rows) with
// async global->LDS b128 copies (contiguous along k), fragments via ds loads.
// The reflection sign is recomputed from x,k-buffers at the epilogue.
// ---------------------------------------------------------------------------
__global__ __launch_bounds__(256) void k_bwd(
    const float* __restrict__ dbuf, const float* __restrict__ W1,
    const float* __restrict__ x, const float* __restrict__ k1,
    const float* __restrict__ k2, const float* __restrict__ k3,
    const float* __restrict__ k4, const float* __restrict__ k5,
    float c1, float c2, float c3, float c4, float c5,
    const float* __restrict__ ctrl, float* __restrict__ kout) {
  __shared__ float As[16][68];
  __shared__ alignas(16) float Bs[128][68];  // Bs[n][k], row stride 272B (16B-aligned)
  int nBase = blockIdx.x * 128;
  int mBase = blockIdx.y * 16;
  int tid = threadIdx.x;
  int wave = tid >> 5;
  int lane = tid & 31;
  int lrow = lane & 15;
  int khalf = lane >> 4;
  int cl = wave * 16 + lrow;
  int col = nBase + cl;
  float dt = ctrl[1];
  v8f acc = {};
  for (int k0 = 0; k0 < HID; k0 += 64) {
    __syncthreads();  // previous iteration's LDS reads complete
    for (int idx = tid; idx < 16 * 64; idx += 256) {
      int r = idx >> 6, c = idx & 63;
      As[r][c] = dbuf[(size_t)(mBase + r) * HID + k0 + c];
    }
    // async-stage W1^T tile: 128 n-rows x 64 k, 16B per lane along k
#pragma unroll
    for (int t2 = 0; t2 < 8; ++t2) {
      int seg = tid + t2 * 256;     // 0..2047
      int n = seg >> 4;             // 0..127
      int kq = (seg & 15) * 4;      // 0..60
      uint32_t lds = (uint32_t)(uintptr_t)&Bs[n][kq];
      const float* gp = &W1[(size_t)(nBase + n) * HID + k0 + kq];
      async_b128(lds, gp);
    }
    wait_async0();
    __syncthreads();
#pragma unroll
    for (int kk = 0; kk < 16; ++kk) {
      int kb = kk * 4 + 2 * khalf;
      v2f a, bf;
      a.x = As[lrow][kb];
      a.y = As[lrow][kb + 1];
      bf.x = Bs[cl][kb];
      bf.y = Bs[cl][kb + 1];
      acc = __builtin_amdgcn_wmma_f32_16x16x4_f32(false, a, false, bf,
                                                  (short)0, acc, false, false);
    }
  }
  int mOff = mBase + khalf * 8;
#pragma unroll
  for (int r = 0; r < 8; ++r) {
    int m = mOff + r;
    size_t gi = (size_t)m * DD + col;
    float zz = x[gi] + dt * (c1 * k1[gi] + c2 * k2[gi] + c3 * k3[gi] +
                             c4 * k4[gi] + c5 * k5[gi]);
    int iv = (int)zz;
    float sgn = ((iv & 1) == 0) ? 1.f : -1.f;
    kout[gi] = -acc[r] * sgn;
  }
}

// ---------------------------------------------------------------------------
// RKF56 combine: x_lo / x_hi, candidate = x_hi, err = max|x_hi - x_lo|
// ---------------------------------------------------------------------------
__global__ __launch_bounds__(256) void k_combine(
    const float* __restrict__ x, const float* __restrict__ k1,
    const float* __restrict__ k3, const float* __restrict__ k4,
    const float* __restrict__ k5, const float* __restrict__ k6,
    float* __restrict__ xcand, const float* __restrict__ ctrl,
    unsigned* __restrict__ errb) {
  __shared__ float red[256];
  int tid = threadIdx.x;
  int i = blockIdx.x * 256 + tid;
  float dt = ctrl[1];
  float lo = x[i] + dt * (25.f / 216.f * k1[i] + 1408.f / 2565.f * k3[i] +
                          2197.f / 4104.f * k4[i] - 0.2f * k5[i]);
  float hi = x[i] + dt * (16.f / 135.f * k1[i] + 6656.f / 12825.f * k3[i] +
                          28561.f / 56430.f * k4[i] - 9.f / 50.f * k5[i] +
                          2.f / 55.f * k6[i]);
  xcand[i] = hi;
  red[tid] = fabsf(hi - lo);
  __syncthreads();
  for (int off = 128; off > 0; off >>= 1) {
    if (tid < off) red[tid] = fmaxf(red[tid], red[tid + off]);
    __syncthreads();
  }
  if (tid == 0) atomicMax(errb, __float_as_uint(red[0]));
}

__global__ void k_control1(float* ctrl, const unsigned* errb) {
  float err = __uint_as_float(errb[0]);
  float t = ctrl[0], dt = ctrl[1];
  bool done = ctrl[2] != 0.f;
  bool accept = (err <= TOLV) || (dt <= MIN_DT + 1e-9f);
  bool upd = accept && !done;
  if (upd) t += dt;
  float grow = 0.9f * dt * powf(TOLV / (err + 1e-12f), 0.2f);
  grow = fminf(fmaxf(grow, MIN_DT), 1.0f);
  float dtn = done ? dt : (accept ? grow : fmaxf(dt * 0.5f, MIN_DT));
  ctrl[0] = t;
  ctrl[1] = dtn;
  ctrl[3] = upd ? 1.f : 0.f;
}

__global__ __launch_bounds__(256) void k_apply(float* __restrict__ x,
                                               const float* __restrict__ xcand,
                                               const float* __restrict__ ctrl) {
  if (ctrl[3] != 0.f) {
    int i = blockIdx.x * 256 + threadIdx.x;
    x[i] = xcand[i];
  }
}

// E[b] = dot(relu_acts[b,:], W2) + b2
__global__ __launch_bounds__(256) void k_energy(const float* __restrict__ abuf,
                                                const float* __restrict__ W2,
                                                const float* __restrict__ b2,
                                                float* __restrict__ E) {
  __shared__ float red[256];
  int b = blockIdx.x, tid = threadIdx.x;
  float s = 0.f;
  for (int j = tid; j < HID; j += 256) s += abuf[(size_t)b * HID + j] * W2[j];
  red[tid] = s;
  __syncthreads();
  for (int off = 128; off > 0; off >>= 1) {
    if (tid < off) red[tid] += red[tid + off];
    __syncthreads();
  }
  if (tid == 0) E[b] = red[0] + b2[0];
}

// Stop test + prevE update (only on accepted active step) + err reset for next step.
__global__ void k_control2(float* ctrl, const float* E, float* prevE,
                           unsigned* errb) {
  errb[0] = 0u;
  if (ctrl[3] == 0.f) return;
  float rel = 0.f;
  for (int b = 0; b < BATCH; ++b) {
    float r = fabsf(E[b] - prevE[b]) / (fabsf(prevE[b]) + 1e-8f);
    rel = fmaxf(rel, r);
    prevE[b] = E[b];
  }
  float t = ctrl[0];
  bool stop = (t > MAX_DELTA_T) || ((t > 0.f) && (rel < REL_TOL));
  if (stop) ctrl[2] = 1.f;
}

// ---------------------------------------------------------------------------
// Final: LayerNorm(ln1) -> logits (1024x10) -> softmax
// ---------------------------------------------------------------------------
__global__ __launch_bounds__(256) void k_final(
    const float* __restrict__ x, const float* __restrict__ g,
    const float* __restrict__ bta, const float* __restrict__ Wo,
    const float* __restrict__ bo, float* __restrict__ out) {
  __shared__ float hs[DD];
  __shared__ float red[256];
  __shared__ float red2[256];
  __shared__ float logits[10];
  int b = blockIdx.x, tid = threadIdx.x;
  float s = 0.f, s2 = 0.f;
  for (int i = tid; i < DD; i += 256) {
    float v = x[(size_t)b * DD + i];
    hs[i] = v;
    s += v;
    s2 += v * v;
  }
  red[tid] = s;
  red2[tid] = s2;
  __syncthreads();
  for (int off = 128; off > 0; off >>= 1) {
    if (tid < off) { red[tid] += red[tid + off]; red2[tid] += red2[tid + off]; }
    __syncthreads();
  }
  float mean = red[0] * (1.0f / DD);
  float var = red2[0] * (1.0f / DD) - mean * mean;
  float rstd = rsqrtf(var + LN_EPS);
  for (int i = tid; i < DD; i += 256)
    hs[i] = (hs[i] - mean) * rstd * g[i] + bta[i];
  __syncthreads();
  for (int j = 0; j < 10; ++j) {
    float p = 0.f;
    for (int i = tid; i < DD; i += 256) p += hs[i] * Wo[(size_t)i * 10 + j];
    red[tid] = p;
    __syncthreads();
    for (int off = 128; off > 0; off >>= 1) {
      if (tid < off) red[tid] += red[tid + off];
      __syncthreads();
    }
    if (tid == 0) logits[j] = red[0] + bo[j];
    __syncthreads();
  }
  if (tid == 0) {
    float mx = logits[0];
    for (int j = 1; j < 10; ++j) mx = fmaxf(mx, logits[j]);
    float se = 0.f, e[10];
    for (int j = 0; j < 10; ++j) { e[j] = expf(logits[j] - mx); se += e[j]; }
    for (int j = 0; j < 10; ++j) out[(size_t)b * 10 + j] = e[j] / se;
  }
}

// ---------------------------------------------------------------------------
extern "C" void kernel_launch(void* const* d_in, const int* in_sizes, int n_in,
                              void* d_out, int out_size, void* d_ws,
                              size_t ws_size, hipStream_t stream) {
  (void)in_sizes; (void)n_in; (void)out_size; (void)ws_size;
  const float* xin   = (const float*)d_in[0];
  const float* ln0_g = (const float*)d_in[1];
  const float* ln0_b = (const float*)d_in[2];
  const float* W1    = (const float*)d_in[3];
  const float* b1    = (const float*)d_in[4];
  const float* W2    = (const float*)d_in[5];
  const float* b2    = (const float*)d_in[6];
  const float* ln1_g = (const float*)d_in[7];
  const float* ln1_b = (const float*)d_in[8];
  const float* Wo    = (const float*)d_in[9];
  const float* bo    = (const float*)d_in[10];
  float* out = (float*)d_out;

  const size_t ND = (size_t)BATCH * DD;
  float* ws = (float*)d_ws;
  float* x  = ws; ws += ND;
  float* xc = ws; ws += ND;
  float* kbuf[6];
  for (int i = 0; i < 6; ++i) { kbuf[i] = ws; ws += ND; }
  float* dbuf  = ws; ws += (size_t)BATCH * HID;
  float* E     = ws; ws += BATCH;
  float* prevE = ws; ws += BATCH;
  float* ctrl  = ws; ws += 8;
  unsigned* errb = (unsigned*)ws;

  static const float CF[6][5] = {
      {0.f, 0.f, 0.f, 0.f, 0.f},
      {0.25f, 0.f, 0.f, 0.f, 0.f},
      {3.f / 32.f, 9.f / 32.f, 0.f, 0.f, 0.f},
      {1932.f / 2197.f, -7200.f / 2197.f, 7296.f / 2197.f, 0.f, 0.f},
      {439.f / 216.f, -8.f, 3680.f / 513.f, -845.f / 4104.f, 0.f},
      {-8.f / 27.f, 2.f, -3544.f / 2565.f, 1859.f / 4104.f, -11.f / 40.f}};

  dim3 blk(256);
  dim3 gElem(ND / 256);                 // 512 blocks, exact
  dim3 gFwd(HID / 128, BATCH / 16);     // 16 x 8
  dim3 gBwd(DD / 128, BATCH / 16);      // 8 x 8

  k_preprocess<<<BATCH, blk, 0, stream>>>(xin, ln0_g, ln0_b, x);
  k_init<<<1, 256, 0, stream>>>(ctrl, errb, prevE);

  for (int step = 0; step < MAX_STEPS; ++step) {
    for (int s = 0; s < 6; ++s) {
      const float* kk[5];
      for (int j = 0; j < 5; ++j) kk[j] = (j < s) ? kbuf[j] : x;  // coef 0 pads
      k_fwd<<<gFwd, blk, 0, stream>>>(x, kk[0], kk[1], kk[2], kk[3], kk[4],
                                      CF[s][0], CF[s][1], CF[s][2], CF[s][3],
                                      CF[s][4], ctrl, W1, b1, W2, dbuf, 0);
      k_bwd<<<gBwd, blk, 0, stream>>>(dbuf, W1, x, kk[0], kk[1], kk[2], kk[3],
                                      kk[4], CF[s][0], CF[s][1], CF[s][2],
                                      CF[s][3], CF[s][4], ctrl, kbuf[s]);
    }
    k_combine<<<gElem, blk, 0, stream>>>(x, kbuf[0], kbuf[2], kbuf[3], kbuf[4],
                                         kbuf[5], xc, ctrl, errb);
    k_control1<<<1, 1, 0, stream>>>(ctrl, errb);
    k_apply<<<gElem, blk, 0, stream>>>(x, xc, ctrl);
    k_fwd<<<gFwd, blk, 0, stream>>>(x, x, x, x, x, x, 0.f, 0.f, 0.f, 0.f, 0.f,
                                    ctrl, W1, b1, W2, dbuf, 1);  // energy acts
    k_energy<<<BATCH, blk, 0, stream>>>(dbuf, W2, b2, E);
    k_control2<<<1, 1, 0, stream>>>(ctrl, E, prevE, errb);
  }
  k_final<<<BATCH, blk, 0, stream>>>(x, ln1_g, ln1_b, Wo, bo, out);
}